// SnapshotGNN_34136400069037
// MI455X (gfx1250) — compile-verified
//
#include <hip/hip_runtime.h>
#include <cstddef>

#define N_IN   128
#define N_HID  128

typedef __attribute__((ext_vector_type(2))) float v2f;
typedef __attribute__((ext_vector_type(8))) float v8f;

// XOR-swizzled LDS layout for a 128x128 f32 weight matrix (64 KB exactly).
// Element (c,k) -> wlds[c*128 + (k ^ ((c&31)<<2))]. Swizzle touches bits >=2,
// so float2 pairs (even k) stay contiguous; the bank index (low 6 bits of the
// DW address) gets c mixed in -> conflict-free b64 reads across 16-lane groups.
#define WSWZ(c, k) (((c) << 7) + ((k) ^ (((c) & 31) << 2)))

// ---------------------------------------------------------------- zero fill
__global__ void sage_zero_f32(float4* __restrict__ p, long n4) {
  long i = (long)blockIdx.x * blockDim.x + threadIdx.x;
  long stride = (long)gridDim.x * blockDim.x;
  for (; i < n4; i += stride) p[i] = make_float4(0.f, 0.f, 0.f, 0.f);
}

// ------------------------------------------------------------- degree count
__global__ void sage_deg_count(const int* __restrict__ dst,
                               float* __restrict__ deg, int E) {
  int e = blockIdx.x * blockDim.x + threadIdx.x;
  if (e < E) atomicAdd(&deg[dst[e]], 1.0f);
}

// deg[n] <- 1 / max(deg[n], 1)  (mean normalization, reused by both layers)
__global__ void sage_inv_deg(float* __restrict__ deg, int N) {
  int n = blockIdx.x * blockDim.x + threadIdx.x;
  if (n < N) deg[n] = 1.0f / fmaxf(deg[n], 1.0f);
}

// --------------------------------------------------- edge scatter (segment_sum)
// One wave32 per edge; each lane moves float4 (128 feats / 32 lanes = 4).
// x (51 MB) and agg (51 MB) both fit in the 192 MB L2, so the gather reads and
// the atomic RMWs resolve at L2 bandwidth, not HBM.
__global__ void sage_scatter(const int* __restrict__ src,
                             const int* __restrict__ dst,
                             const float* __restrict__ x,
                             float* __restrict__ agg, int E) {
  int gid  = blockIdx.x * blockDim.x + threadIdx.x;
  int wid  = gid >> 5;          // edge id
  int lane = threadIdx.x & 31;
  if (wid >= E) return;
  int s = src[wid];
  int d = dst[wid];
  const float4 v = *(const float4*)(x + (long)s * N_IN + lane * 4);
  float* ap = agg + (long)d * N_IN + lane * 4;
  atomicAdd(ap + 0, v.x);
  atomicAdd(ap + 1, v.y);
  atomicAdd(ap + 2, v.z);
  atomicAdd(ap + 3, v.w);
}

// -------------------------------------------- cooperative LDS weight staging
__device__ __forceinline__ void sage_load_w(float* __restrict__ wlds,
                                            const float* __restrict__ W,
                                            int tid) {
  const float2* src = (const float2*)W;
  #pragma unroll 4
  for (int j = tid; j < 128 * 64; j += 256) {
    int c  = j >> 6;
    int k2 = (j & 63) << 1;
    float2 v = src[j];
    *(float2*)&wlds[WSWZ(c, k2)] = v;
  }
}

// ------------------------------------------------------------------- SAGE GEMM
// out_block = ReLU( X @ Wl^T + (AGG*invd) @ Wr^T + bl )
// Block = 256 threads (8 waves); wave w owns rows [blk*128 + 16w, +16) and
// ALL 128 output columns as 8 WMMA f32 accumulator tiles (64 VGPRs).
// HEAD=false: store H.  HEAD=true: fuse output head  out = h2 @ Wo^T + bo.
template <bool HEAD>
__global__ void __launch_bounds__(256)
sage_gemm(const float* __restrict__ X, const float* __restrict__ AGG,
          const float* __restrict__ invd,
          const float* __restrict__ Wl, const float* __restrict__ Wr,
          const float* __restrict__ bl,
          const float* __restrict__ Wo, const float* __restrict__ bo,
          float* __restrict__ H, float* __restrict__ OUT, int N) {
  __shared__ float wlds[128 * 128];  // 64 KB, reused for Wl then Wr

  const int tid  = threadIdx.x;
  const int lane = tid & 31;
  const int wv   = tid >> 5;
  const int half = lane >> 4;   // 0: K={0,1}, 1: K={2,3} (f32 A-operand layout)
  const int l16  = lane & 15;

  const int  row0   = blockIdx.x * 128 + wv * 16;
  int        myrow  = row0 + l16;
  const int  rclamp = myrow < N ? myrow : (N - 1);  // keep EXEC all-ones for WMMA
  const float inv   = invd[rclamp];

  const float* Xp = X   + (long)rclamp * N_IN;
  const float* Mp = AGG + (long)rclamp * N_IN;

  // Pull the pass-2 A row (4 cachelines) toward the caches while pass 1 runs.
  __builtin_prefetch(Mp +  0, 0, 1);   // global_prefetch_b8
  __builtin_prefetch(Mp + 32, 0, 1);
  __builtin_prefetch(Mp + 64, 0, 1);
  __builtin_prefetch(Mp + 96, 0, 1);

  v8f acc[8] = {};  // 16x16 f32 tiles, columns [16t, 16t+16)

  // ---- pass 1: self term  X @ Wl^T
  sage_load_w(wlds, Wl, tid);
  __syncthreads();
  #pragma unroll 2
  for (int kk = 0; kk < 32; ++kk) {
    const int kl = (kk << 2) + (half << 1);
    v2f a = *(const v2f*)(Xp + kl);
    #pragma unroll
    for (int t = 0; t < 8; ++t) {
      const int c = l16 + (t << 4);
      v2f b = *(const v2f*)&wlds[WSWZ(c, kl)];
      acc[t] = __builtin_amdgcn_wmma_f32_16x16x4_f32(
          false, a, false, b, (short)0, acc[t], false, false);
    }
  }
  __syncthreads();

  // ---- pass 2: neighbor-mean term  (AGG * invd) @ Wr^T  (fused normalization)
  sage_load_w(wlds, Wr, tid);
  __syncthreads();
  #pragma unroll 2
  for (int kk = 0; kk < 32; ++kk) {
    const int kl = (kk << 2) + (half << 1);
    v2f a = *(const v2f*)(Mp + kl);
    a.x *= inv;
    a.y *= inv;
    #pragma unroll
    for (int t = 0; t < 8; ++t) {
      const int c = l16 + (t << 4);
      v2f b = *(const v2f*)&wlds[WSWZ(c, kl)];
      acc[t] = __builtin_amdgcn_wmma_f32_16x16x4_f32(
          false, a, false, b, (short)0, acc[t], false, false);
    }
  }

  // ---- epilogue. C/D tile layout: lane l16 = column within tile,
  //      VGPR r -> row  row0 + r + 8*half.
  if (!HEAD) {
    #pragma unroll
    for (int t = 0; t < 8; ++t) {
      const int c  = l16 + (t << 4);
      const float bb = bl[c];
      #pragma unroll
      for (int r = 0; r < 8; ++r) {
        const int m = row0 + r + (half << 3);
        if (m < N) {
          float v = acc[t][r] + bb;
          H[(long)m * N_HID + c] = v > 0.f ? v : 0.f;
        }
      }
    }
  } else {
    // out[m] = sum_c ReLU(acc + bl2)[m][c] * Wo[c] + bo
    float s[8] = {0.f, 0.f, 0.f, 0.f, 0.f, 0.f, 0.f, 0.f};
    #pragma unroll
    for (int t = 0; t < 8; ++t) {
      const int c  = l16 + (t << 4);
      const float bb = bl[c];
      const float wo = Wo[c];
      #pragma unroll
      for (int r = 0; r < 8; ++r) {
        float v = acc[t][r] + bb;
        v = v > 0.f ? v : 0.f;
        s[r] += v * wo;
      }
    }
    // reduce across the 16-lane half-groups (xor masks 1,2,4,8 stay in-group)
    #pragma unroll
    for (int off = 1; off < 16; off <<= 1) {
      #pragma unroll
      for (int r = 0; r < 8; ++r) s[r] += __shfl_xor(s[r], off, 32);
    }
    if (l16 == 0) {
      const float b0 = bo[0];
      #pragma unroll
      for (int r = 0; r < 8; ++r) {
        const int m = row0 + r + (half << 3);
        if (m < N) OUT[m] = s[r] + b0;
      }
    }
  }
}

// ------------------------------------------------------------------ launcher
extern "C" void kernel_launch(void* const* d_in, const int* in_sizes, int n_in,
                              void* d_out, int out_size, void* d_ws, size_t ws_size,
                              hipStream_t stream) {
  const float* x   = (const float*)d_in[0];
  const int*   ei  = (const int*)d_in[1];
  const float* Wl1 = (const float*)d_in[2];
  const float* bl1 = (const float*)d_in[3];
  const float* Wr1 = (const float*)d_in[4];
  const float* Wl2 = (const float*)d_in[5];
  const float* bl2 = (const float*)d_in[6];
  const float* Wr2 = (const float*)d_in[7];
  const float* Wo  = (const float*)d_in[8];
  const float* bo  = (const float*)d_in[9];
  float* out = (float*)d_out;

  const int N = in_sizes[0] / N_IN;
  const int E = in_sizes[1] / 2;
  const int* src = ei;
  const int* dst = ei + E;

  // workspace: deg(N) | agg(N*128) | h(N*128)   (~103 MB total)
  char* w = (char*)d_ws;
  size_t o = 0;
  float* deg = (float*)(w + o); o += ((size_t)N * 4 + 1023) & ~(size_t)1023;
  float* agg = (float*)(w + o); o += (size_t)N * N_IN * 4;
  float* h   = (float*)(w + o);

  const long nfeat  = (long)N * N_IN;
  const int  zb_deg = (int)((N / 4 + 255) / 256);
  const int  zb_agg = (int)((nfeat / 4 + 255) / 256);
  const int  eb     = (E + 255) / 256;
  const int  nb     = (N + 255) / 256;
  const int  sb     = (E * 32 + 255) / 256;   // one wave per edge
  const int  gb     = (N + 127) / 128;        // 128 rows per block (8 waves x 16)

  // degree (shared by both layers)
  sage_zero_f32<<<zb_deg, 256, 0, stream>>>((float4*)deg, N / 4);
  sage_deg_count<<<eb, 256, 0, stream>>>(dst, deg, E);
  sage_inv_deg<<<nb, 256, 0, stream>>>(deg, N);

  // layer 1
  sage_zero_f32<<<zb_agg, 256, 0, stream>>>((float4*)agg, nfeat / 4);
  sage_scatter<<<sb, 256, 0, stream>>>(src, dst, x, agg, E);
  sage_gemm<false><<<gb, 256, 0, stream>>>(x, agg, deg, Wl1, Wr1, bl1,
                                           nullptr, nullptr, h, nullptr, N);

  // layer 2 (+ fused output head)
  sage_zero_f32<<<zb_agg, 256, 0, stream>>>((float4*)agg, nfeat / 4);
  sage_scatter<<<sb, 256, 0, stream>>>(src, dst, h, agg, E);
  sage_gemm<true><<<gb, 256, 0, stream>>>(h, agg, deg, Wl2, Wr2, bl2,
                                          Wo, bo, nullptr, out, N);
}